// Net_71605694759738
// MI455X (gfx1250) — compile-verified
//
#include <hip/hip_runtime.h>

#define NREL 6
#define NBASES 4
#define DIM 32
#define NW 7  // 6 relations + root matrix as pseudo-relation 6

typedef __attribute__((ext_vector_type(2))) float v2f;
typedef __attribute__((ext_vector_type(8))) float v8f;

// ---------------------------------------------------------------------------
// W[w][k][n] (row-major 32x32 each): w<6 -> sum_b comp[w,b]*basis[b], w==6 -> root
// ---------------------------------------------------------------------------
__global__ void build_w_kernel(const float* __restrict__ comp,
                               const float* __restrict__ basis,
                               const float* __restrict__ root,
                               float* __restrict__ W) {
    int idx = blockIdx.x * blockDim.x + threadIdx.x;
    if (idx >= NW * DIM * DIM) return;
    int w  = idx / (DIM * DIM);
    int ko = idx % (DIM * DIM);
    float acc;
    if (w < NREL) {
        acc = 0.0f;
        #pragma unroll
        for (int b = 0; b < NBASES; ++b)
            acc += comp[w * NBASES + b] * basis[b * DIM * DIM + ko];
    } else {
        acc = root[ko];
    }
    W[idx] = acc;
}

// ---------------------------------------------------------------------------
// h[w][node][out] = x[node][:] @ W[w][:][:]   via V_WMMA_F32_16X16X4_F32
// One wave per (16-node tile, w): two 16x16 D tiles (out cols 0-15 / 16-31),
// K = 32 consumed as 8 steps of K=4.
// A frag (16x4 f32, 2 VGPR): lane<16 -> K={0,1}, lane>=16 -> K={2,3}, M=lane&15
// B frag (4x16 f32, 2 VGPR): same K split, N=lane&15
// C/D (16x16 f32, 8 VGPR):   vgpr v -> row v (lane<16) / v+8 (lane>=16), col=lane&15
// ---------------------------------------------------------------------------
__global__ void transform_kernel(const float* __restrict__ x,
                                 const float* __restrict__ W,
                                 float* __restrict__ h,
                                 int n_tiles, int n_nodes) {
    int wave = threadIdx.x >> 5;
    int lane = threadIdx.x & 31;
    int tile = blockIdx.x * (blockDim.x >> 5) + wave;
    int w    = blockIdx.y;
    if (tile >= n_tiles) return;  // wave-uniform

    int m  = lane & 15;
    int kp = lane >> 4;  // which K-pair this lane supplies (0 or 1)
    int row = tile * 16 + m;
    int arow = row < n_nodes ? row : (n_nodes - 1);  // clamp A loads; stores guarded
    const float* xrow = x + (size_t)arow * DIM;
    const float* Wm   = W + (size_t)w * DIM * DIM;

    v8f c0 = {};
    v8f c1 = {};
    #pragma unroll
    for (int ks = 0; ks < 8; ++ks) {
        int k0 = ks * 4 + kp * 2;
        v2f a;
        a.x = xrow[k0];
        a.y = xrow[k0 + 1];
        v2f b0, b1;
        b0.x = Wm[k0 * DIM + m];
        b0.y = Wm[(k0 + 1) * DIM + m];
        b1.x = Wm[k0 * DIM + m + 16];
        b1.y = Wm[(k0 + 1) * DIM + m + 16];
        c0 = __builtin_amdgcn_wmma_f32_16x16x4_f32(false, a, false, b0,
                                                   (short)0, c0, false, false);
        c1 = __builtin_amdgcn_wmma_f32_16x16x4_f32(false, a, false, b1,
                                                   (short)0, c1, false, false);
    }

    float* hw = h + (size_t)w * n_nodes * DIM;
    int mbase = tile * 16 + (kp ? 8 : 0);
    #pragma unroll
    for (int v = 0; v < 8; ++v) {
        int r = mbase + v;
        if (r < n_nodes) {
            hw[(size_t)r * DIM + m]      = c0[v];
            hw[(size_t)r * DIM + m + 16] = c1[v];
        }
    }
}

// ---------------------------------------------------------------------------
// cnt[dst*NREL + rel] += 1  (graph-structure counts; same for both layers)
// ---------------------------------------------------------------------------
__global__ void count_kernel(const int* __restrict__ dst,
                             const int* __restrict__ etype,
                             float* __restrict__ cnt, int n_edges) {
    int e = blockIdx.x * blockDim.x + threadIdx.x;
    if (e >= n_edges) return;
    atomicAdd(&cnt[(size_t)dst[e] * NREL + etype[e]], 1.0f);
}

// ---------------------------------------------------------------------------
// Wave-per-edge scatter: lane = feature dim (coalesced 128B atomic footprint).
// h is L2-resident (89.6 MB < 192 MB L2) so the random gather hits L2.
// ---------------------------------------------------------------------------
__global__ void scatter_kernel(const float* __restrict__ h,
                               const int* __restrict__ src,
                               const int* __restrict__ dst,
                               const int* __restrict__ etype,
                               float* __restrict__ sums,
                               int n_edges, int n_nodes) {
    long long gid = (long long)blockIdx.x * blockDim.x + threadIdx.x;
    int e    = (int)(gid >> 5);
    int lane = (int)(gid & 31);
    if (e >= n_edges) return;
    int r = etype[e];
    int s = src[e];
    int d = dst[e];
    float v = h[((size_t)r * n_nodes + s) * DIM + lane];
    atomicAdd(&sums[((size_t)d * NREL + r) * DIM + lane], v);
}

// ---------------------------------------------------------------------------
// out[n][d] = sum_r sums[n,r,d]/max(cnt[n,r],1) + hroot[n][d] + bias[d] (+ReLU)
// ---------------------------------------------------------------------------
__global__ void finalize_kernel(const float* __restrict__ sums,
                                const float* __restrict__ cnt,
                                const float* __restrict__ hroot,
                                const float* __restrict__ bias,
                                float* __restrict__ out,
                                int n_nodes, int do_relu) {
    long long idx = (long long)blockIdx.x * blockDim.x + threadIdx.x;
    if (idx >= (long long)n_nodes * DIM) return;
    int n = (int)(idx >> 5);
    int d = (int)(idx & 31);
    float acc = hroot[(size_t)n * DIM + d] + bias[d];
    #pragma unroll
    for (int r = 0; r < NREL; ++r) {
        float c = cnt[(size_t)n * NREL + r];
        acc += sums[((size_t)n * NREL + r) * DIM + d] / fmaxf(c, 1.0f);
    }
    if (do_relu) acc = fmaxf(acc, 0.0f);
    out[idx] = acc;
}

// ---------------------------------------------------------------------------
extern "C" void kernel_launch(void* const* d_in, const int* in_sizes, int n_in,
                              void* d_out, int out_size, void* d_ws, size_t ws_size,
                              hipStream_t stream) {
    (void)n_in; (void)out_size; (void)ws_size;

    const float* embedding  = (const float*)d_in[0];
    const int*   edge_index = (const int*)d_in[1];
    const int*   edge_type  = (const int*)d_in[2];
    const float* comp1  = (const float*)d_in[3];
    const float* basis1 = (const float*)d_in[4];
    const float* root1  = (const float*)d_in[5];
    const float* bias1  = (const float*)d_in[6];
    const float* comp2  = (const float*)d_in[7];
    const float* basis2 = (const float*)d_in[8];
    const float* root2  = (const float*)d_in[9];
    const float* bias2  = (const float*)d_in[10];

    const int n_nodes = in_sizes[0] / DIM;
    const int n_edges = in_sizes[2];
    const int* src = edge_index;              // edge_index[0, :]
    const int* dst = edge_index + n_edges;    // edge_index[1, :]

    // ---- workspace carve (256B aligned slabs) ----
    char*  ws  = (char*)d_ws;
    size_t off = 0;
    auto alloc_f = [&](size_t nfloats) -> float* {
        float* p = (float*)(ws + off);
        off += ((nfloats * sizeof(float)) + 255) & ~(size_t)255;
        return p;
    };
    float* Wbuf  = alloc_f((size_t)NW * DIM * DIM);            //  28 KB
    float* cnt   = alloc_f((size_t)n_nodes * NREL);            // 2.4 MB
    float* sums  = alloc_f((size_t)n_nodes * NREL * DIM);      //  77 MB
    float* hbuf  = alloc_f((size_t)NW * n_nodes * DIM);        //  90 MB
    float* x_mid = alloc_f((size_t)n_nodes * DIM);             //  13 MB
    float* hroot = hbuf + (size_t)NREL * n_nodes * DIM;        // pseudo-rel 6

    const int n_tiles = (n_nodes + 15) / 16;
    dim3 tblock(256);
    dim3 tgrid((n_tiles + 7) / 8, NW);
    const int wblocks  = (NW * DIM * DIM + 255) / 256;
    const int cblocks  = (n_edges + 255) / 256;
    const long long sthreads = (long long)n_edges * 32;
    const int sblocks  = (int)((sthreads + 255) / 256);
    const long long fthreads = (long long)n_nodes * DIM;
    const int fblocks  = (int)((fthreads + 255) / 256);

    // graph counts (recomputed every call for determinism)
    hipMemsetAsync(cnt, 0, (size_t)n_nodes * NREL * sizeof(float), stream);
    count_kernel<<<cblocks, 256, 0, stream>>>(dst, edge_type, cnt, n_edges);

    // ---------------- layer 1 ----------------
    build_w_kernel<<<wblocks, 256, 0, stream>>>(comp1, basis1, root1, Wbuf);
    transform_kernel<<<tgrid, tblock, 0, stream>>>(embedding, Wbuf, hbuf,
                                                   n_tiles, n_nodes);
    hipMemsetAsync(sums, 0, (size_t)n_nodes * NREL * DIM * sizeof(float), stream);
    scatter_kernel<<<sblocks, 256, 0, stream>>>(hbuf, src, dst, edge_type,
                                                sums, n_edges, n_nodes);
    finalize_kernel<<<fblocks, 256, 0, stream>>>(sums, cnt, hroot, bias1,
                                                 x_mid, n_nodes, /*relu=*/1);

    // ---------------- layer 2 ----------------
    build_w_kernel<<<wblocks, 256, 0, stream>>>(comp2, basis2, root2, Wbuf);
    transform_kernel<<<tgrid, tblock, 0, stream>>>(x_mid, Wbuf, hbuf,
                                                   n_tiles, n_nodes);
    hipMemsetAsync(sums, 0, (size_t)n_nodes * NREL * DIM * sizeof(float), stream);
    scatter_kernel<<<sblocks, 256, 0, stream>>>(hbuf, src, dst, edge_type,
                                                sums, n_edges, n_nodes);
    finalize_kernel<<<fblocks, 256, 0, stream>>>(sums, cnt, hroot, bias2,
                                                 (float*)d_out, n_nodes, /*relu=*/0);
}